// lightweight_Convolution_21320217657915
// MI455X (gfx1250) — compile-verified
//
#include <hip/hip_runtime.h>
#include <hip/hip_bf16.h>

typedef float v2f __attribute__((ext_vector_type(2)));
typedef float v8f __attribute__((ext_vector_type(8)));

#define BN 16
#define SN 4096
#define DN 1024
#define HN 16
#define KN 15
#define DK 64

#define WAVES_PER_BLOCK 8
#define TILES_PER_WAVE  4
// rows per tile = 16 ; rows per block = 8*4*16 = 512 (divides S=4096 -> one head per block)

__global__ __launch_bounds__(256)
void dwconv_wmma_f32_kernel(const float* __restrict__ x,
                            const float* __restrict__ W,
                            const float* __restrict__ bvec,
                            float* __restrict__ out)
{
    __shared__ float sk[KN + 1];   // softmaxed kernel taps + bias

    const int tid  = threadIdx.x;
    const int lane = tid & 31;
    const int wave = tid >> 5;

    const long long blockTile0 = (long long)blockIdx.x * (WAVES_PER_BLOCK * TILES_PER_WAVE);
    const long long r0_block   = blockTile0 * 16;            // first row of this block
    const int h = (int)((r0_block / SN) % HN);
    const int b = (int)( r0_block / ((long long)SN * HN));

    // ---- softmax of W[h, 0..14] once per block ----
    if (tid == 0) {
        float w[KN];
        float m = -3.4e38f;
        #pragma unroll
        for (int j = 0; j < KN; ++j) { w[j] = W[h * KN + j]; m = fmaxf(m, w[j]); }
        float s = 0.f;
        #pragma unroll
        for (int j = 0; j < KN; ++j) { w[j] = __expf(w[j] - m); s += w[j]; }
        const float inv = 1.0f / s;
        #pragma unroll
        for (int j = 0; j < KN; ++j) sk[j] = w[j] * inv;
        sk[KN] = bvec[h];
    }
    __syncthreads();

    const int hi = lane >> 4;      // 0: lanes 0-15, 1: lanes 16-31
    const int nn = lane & 15;
    const float bh = sk[KN];

    // ---- 8 banded B fragments: element v holds T[4*kb + v + 2*hi, 16*n + nn],
    //      depends only on d = 4*kb - 16*n = 4*bi - 8 ----
    v2f bfrag[8];
    #pragma unroll
    for (int bi = 0; bi < 8; ++bi) {
        const int d = 4 * bi - 8;
        #pragma unroll
        for (int v = 0; v < 2; ++v) {
            const int kk  = v + 2 * hi;
            const int idx = d + kk - nn + 7;      // tap index into kern
            bfrag[bi][v] = (idx >= 0 && idx < KN) ? sk[idx] : 0.0f;
        }
    }

    const long long waveTile0 = blockTile0 + (long long)wave * TILES_PER_WAVE;

    for (int t = 0; t < TILES_PER_WAVE; ++t) {
        const long long tile = waveTile0 + t;
        const long long r0   = tile * 16;                    // first input row of tile
        const int s0         = (int)(r0 % SN);

        const float* xt = x + r0 * (long long)DK;

        if (t + 1 < TILES_PER_WAVE) {
            __builtin_prefetch(xt + 16 * DK + (size_t)nn * DK, 0, 1);
        }

        // ---- A fragments: a[kb] = X[nn, 4*kb + 2*hi .. +1] (16x4 f32 WMMA layout) ----
        v2f a[16];
        #pragma unroll
        for (int kb = 0; kb < 16; ++kb) {
            a[kb] = *(const v2f*)(xt + nn * DK + 4 * kb + 2 * hi);
        }

        float* orow = out + (((long long)h * BN + b) * SN + s0) * (long long)DK;

        // ---- 4 output col-blocks of 16; band clipping via bi range ----
        #pragma unroll
        for (int n = 0; n < 4; ++n) {
            v8f acc = {};
            #pragma unroll
            for (int kb = 0; kb < 16; ++kb) {
                const int bi = kb - 4 * n + 2;               // fragment selector
                if (bi >= 0 && bi < 8) {
                    acc = __builtin_amdgcn_wmma_f32_16x16x4_f32(
                        /*neg_a=*/false, a[kb],
                        /*neg_b=*/false, bfrag[bi],
                        /*c_mod=*/(short)0, acc,
                        /*reuse_a=*/false, /*reuse_b=*/false);
                }
            }
            // bias + ReLU + store: element v -> (M = v + 8*hi, N = 16*n + nn)
            #pragma unroll
            for (int v = 0; v < 8; ++v) {
                const float val = fmaxf(acc[v] + bh, 0.0f);
                orow[(long long)(v + 8 * hi) * DK + 16 * n + nn] = val;
            }
        }
    }
}

extern "C" void kernel_launch(void* const* d_in, const int* in_sizes, int n_in,
                              void* d_out, int out_size, void* d_ws, size_t ws_size,
                              hipStream_t stream) {
    const float* x  = (const float*)d_in[0];
    const float* W  = (const float*)d_in[1];
    const float* bv = (const float*)d_in[2];
    float* out = (float*)d_out;

    const long long totalRows  = (long long)BN * HN * SN;            // 1,048,576
    const long long totalTiles = totalRows / 16;                     // 65,536
    const int rowsPerBlock     = WAVES_PER_BLOCK * TILES_PER_WAVE;   // 32 tiles
    const int grid             = (int)(totalTiles / rowsPerBlock);   // 2048

    dwconv_wmma_f32_kernel<<<grid, 256, 0, stream>>>(x, W, bv, out);
}